// GraphVectorEncoder_11321533792935
// MI455X (gfx1250) — compile-verified
//
#include <hip/hip_runtime.h>
#include <hip/hip_bf16.h>
#include <cstdint>

// ---------------------------------------------------------------------------
// Problem constants (match reference)
// ---------------------------------------------------------------------------
#define NNODES 50000
#define NEDGES 800000
#define NGRAPH 64
#define DMODEL 128   // all layers: din = dout = 128

typedef float v2f __attribute__((ext_vector_type(2)));
typedef float v8f __attribute__((ext_vector_type(8)));

// Monotone order-preserving encoding of float -> uint for atomicMax
__device__ __forceinline__ unsigned enc_f32(float f) {
    unsigned u = __float_as_uint(f);
    return (u & 0x80000000u) ? ~u : (u | 0x80000000u);
}
__device__ __forceinline__ float dec_f32(unsigned k) {
    unsigned u = (k & 0x80000000u) ? (k & 0x7FFFFFFFu) : ~k;
    return __uint_as_float(u);
}
#define ENC_NEG_INF 0x007FFFFFu   // enc(-inf)

// ---------------------------------------------------------------------------
// Fill helpers (d_ws is poisoned; we must zero accumulators every call)
// ---------------------------------------------------------------------------
__global__ void fill_f32(float* __restrict__ p, float v, int n) {
    int t = blockIdx.x * blockDim.x + threadIdx.x;
    if (t < n) p[t] = v;
}
__global__ void fill_u32(unsigned* __restrict__ p, unsigned v, int n) {
    int t = blockIdx.x * blockDim.x + threadIdx.x;
    if (t < n) p[t] = v;
}

// ---------------------------------------------------------------------------
// Fused Q/K/V/S GEMM: O = A[50000x128] * W[128x128] + b, via fp32 WMMA.
// grid = (3125 row-tiles, 4 matrices), block = 256 (8 waves).
// Wave w computes output tile rows [16*bx .. +16), cols [16*w .. +16).
// fp32 WMMA fragment layouts (ISA 7.12.2):
//   A 16x4 : lane&15 = row M; v[0]=A[M][k0+2*hi], v[1]=A[M][k0+2*hi+1], hi=lane>>4
//   B 4x16 : lane&15 = col N; v[0]=B[k0+2*hi][N], v[1]=B[k0+2*hi+1][N]
//   C 16x16: VGPR i holds row (i + 8*hi), col = lane&15
// ---------------------------------------------------------------------------
__global__ void __launch_bounds__(256)
gemm_qkvs_wmma(const float* __restrict__ A,
               const float* __restrict__ Wq, const float* __restrict__ Wk,
               const float* __restrict__ Wv, const float* __restrict__ Ws,
               const float* __restrict__ bq, const float* __restrict__ bk,
               const float* __restrict__ bv, const float* __restrict__ bs,
               float* __restrict__ Q, float* __restrict__ K,
               float* __restrict__ V, float* __restrict__ S) {
    const float* W;
    const float* b;
    float* O;
    switch (blockIdx.y) {
        case 0:  W = Wq; b = bq; O = Q; break;
        case 1:  W = Wk; b = bk; O = K; break;
        case 2:  W = Wv; b = bv; O = V; break;
        default: W = Ws; b = bs; O = S; break;
    }
    const int wave = threadIdx.x >> 5;      // 0..7 -> column tile
    const int lane = threadIdx.x & 31;
    const int hi   = lane >> 4;             // half-wave select
    const int lid  = lane & 15;
    const int rowBase = blockIdx.x * 16;    // 3125*16 == 50000 exactly
    const int colBase = wave * 16;
    const int r = rowBase + lid;            // A-fragment row
    const int c = colBase + lid;            // B/C-fragment col

    v8f acc = {};
#pragma unroll
    for (int k0 = 0; k0 < DMODEL; k0 += 4) {
        const int kk = k0 + 2 * hi;
        v2f a, bb;
        a.x  = A[(size_t)r * DMODEL + kk];          // contiguous pair -> b64 load
        a.y  = A[(size_t)r * DMODEL + kk + 1];
        bb.x = W[(size_t)kk * DMODEL + c];
        bb.y = W[(size_t)(kk + 1) * DMODEL + c];
        acc = __builtin_amdgcn_wmma_f32_16x16x4_f32(
                  /*neg_a=*/false, a, /*neg_b=*/false, bb,
                  /*c_mod=*/(short)0, acc, /*reuse_a=*/false, /*reuse_b=*/false);
    }

    const float bias = b[c];
#pragma unroll
    for (int i = 0; i < 8; ++i) {
        const int row = rowBase + i + 8 * hi;
        O[(size_t)row * DMODEL + c] = acc[i] + bias;
    }
}

// ---------------------------------------------------------------------------
// Per-(edge,head): alpha = q[dst]·k[src] / sqrt(hd); segment max via atomicMax.
// ---------------------------------------------------------------------------
__global__ void __launch_bounds__(256)
edge_alpha(const long long* __restrict__ src, const long long* __restrict__ dst,
           const float* __restrict__ Q, const float* __restrict__ K,
           float* __restrict__ alpha, unsigned* __restrict__ amax,
           int H, float inv_sqrt_hd) {
    const int t = blockIdx.x * blockDim.x + threadIdx.x;
    if (t >= NEDGES * H) return;
    const int e = t / H;
    const int h = t - e * H;
    const int hd = DMODEL / H;
    const int sN = (int)src[e];
    const int dN = (int)dst[e];
    const float4* qp = (const float4*)(Q + (size_t)dN * DMODEL + h * hd);
    const float4* kp = (const float4*)(K + (size_t)sN * DMODEL + h * hd);
    float acc = 0.0f;
    for (int j = 0; j < hd / 4; ++j) {
        float4 a = qp[j];
        float4 b = kp[j];
        acc += a.x * b.x + a.y * b.y + a.z * b.z + a.w * b.w;
    }
    acc *= inv_sqrt_hd;
    alpha[t] = acc;
    atomicMax(&amax[(size_t)dN * H + h], enc_f32(acc));
}

// ---------------------------------------------------------------------------
// Wave per edge: p = exp(alpha - amax[dst]); denom += p; agg[dst] += p * v[src].
// Each lane covers 4 of the 128 feature dims.
// ---------------------------------------------------------------------------
__global__ void __launch_bounds__(256)
edge_msg(const long long* __restrict__ src, const long long* __restrict__ dst,
         const float* __restrict__ V, const float* __restrict__ alpha,
         const unsigned* __restrict__ amax,
         float* __restrict__ denom, float* __restrict__ agg, int H) {
    const int wid  = (blockIdx.x * blockDim.x + threadIdx.x) >> 5;  // edge id
    const int lane = threadIdx.x & 31;
    if (wid >= NEDGES) return;
    const int sN = (int)src[wid];
    const int dN = (int)dst[wid];

    float p[2];
#pragma unroll
    for (int h = 0; h < 2; ++h) {
        if (h < H) {
            const float m = dec_f32(amax[(size_t)dN * H + h]);
            p[h] = __expf(alpha[(size_t)wid * H + h] - m);
        } else {
            p[h] = 0.0f;
        }
    }
    if (lane < H) atomicAdd(&denom[(size_t)dN * H + lane], p[lane]);

#pragma unroll
    for (int i = 0; i < 4; ++i) {
        const int d = lane + 32 * i;
        const int h = (d * H) >> 7;        // d / (128/H)
        const float val = p[h] * V[(size_t)sN * DMODEL + d];
        atomicAdd(&agg[(size_t)dN * DMODEL + d], val);
    }
}

// ---------------------------------------------------------------------------
// out = relu(agg/denom + skip), in place on agg buffer.
// ---------------------------------------------------------------------------
__global__ void __launch_bounds__(256)
finalize_node(const float* __restrict__ S, const float* __restrict__ denom,
              float* __restrict__ hout, int H) {
    const int t = blockIdx.x * blockDim.x + threadIdx.x;
    if (t >= NNODES * DMODEL) return;
    const int node = t >> 7;
    const int d    = t & 127;
    const int h    = (d * H) >> 7;
    const float dn = denom[(size_t)node * H + h] + 1e-16f;
    const float v  = hout[t] / dn + S[t];
    hout[t] = v > 0.0f ? v : 0.0f;
}

// ---------------------------------------------------------------------------
// Global mean pool over graphs.
// ---------------------------------------------------------------------------
__global__ void __launch_bounds__(256)
pool_acc(const float* __restrict__ h, const long long* __restrict__ batch,
         float* __restrict__ sums, float* __restrict__ cnts) {
    const int t = blockIdx.x * blockDim.x + threadIdx.x;
    if (t >= NNODES * DMODEL) return;
    const int node = t >> 7;
    const int d    = t & 127;
    const int g    = (int)batch[node];
    atomicAdd(&sums[(size_t)g * DMODEL + d], h[t]);
    if (d == 0) atomicAdd(&cnts[g], 1.0f);
}

__global__ void __launch_bounds__(256)
pool_div(const float* __restrict__ sums, const float* __restrict__ cnts,
         float* __restrict__ out) {
    const int t = blockIdx.x * blockDim.x + threadIdx.x;
    if (t >= NGRAPH * DMODEL) return;
    const int g = t >> 7;
    out[t] = sums[t] / fmaxf(cnts[g], 1.0f);
}

// ---------------------------------------------------------------------------
// Host side
// ---------------------------------------------------------------------------
extern "C" void kernel_launch(void* const* d_in, const int* in_sizes, int n_in,
                              void* d_out, int out_size, void* d_ws, size_t ws_size,
                              hipStream_t stream) {
    (void)in_sizes; (void)n_in; (void)out_size; (void)ws_size;

    const float*     x     = (const float*)d_in[0];
    const long long* ei    = (const long long*)d_in[1];   // int64 [2, E]
    const long long* srcI  = ei;
    const long long* dstI  = ei + NEDGES;
    const long long* batch = (const long long*)d_in[2];   // int64 [N]

    // Workspace carve-up
    const size_t NF = (size_t)NNODES * DMODEL;            // 6.4M floats
    float*    hA    = (float*)d_ws;
    float*    hB    = hA + NF;
    float*    q     = hB + NF;
    float*    k     = q  + NF;
    float*    v     = k  + NF;
    float*    s     = v  + NF;
    float*    alpha = s  + NF;                            // E * 2
    float*    denom = alpha + (size_t)NEDGES * 2;         // N * 2
    unsigned* amax  = (unsigned*)(denom + (size_t)NNODES * 2);  // N * 2
    float*    psum  = (float*)(amax + (size_t)NNODES * 2);      // G * 128
    float*    pcnt  = psum + (size_t)NGRAPH * DMODEL;           // G

    const int   heads[3] = {2, 2, 1};
    float*      outs[3]  = {hA, hB, hA};
    const float* in      = x;

    const int B256      = 256;
    const int gNF       = (int)((NF + 255) / 256);          // node*feat grid
    const int gEdgeWave = (NEDGES + 7) / 8;                 // wave-per-edge grid

    for (int li = 0; li < 3; ++li) {
        const int base = 3 + 8 * li;
        const float* Wq = (const float*)d_in[base + 0];
        const float* Wk = (const float*)d_in[base + 1];
        const float* Wv = (const float*)d_in[base + 2];
        const float* Ws = (const float*)d_in[base + 3];
        const float* bq = (const float*)d_in[base + 4];
        const float* bk = (const float*)d_in[base + 5];
        const float* bv = (const float*)d_in[base + 6];
        const float* bs = (const float*)d_in[base + 7];

        const int   H   = heads[li];
        const int   hd  = DMODEL / H;
        float*      out = outs[li];

        // 1) Q/K/V/Skip projections via fp32 WMMA
        gemm_qkvs_wmma<<<dim3(NNODES / 16, 4), B256, 0, stream>>>(
            in, Wq, Wk, Wv, Ws, bq, bk, bv, bs, q, k, v, s);

        // 2) Reset segment accumulators
        fill_f32<<<gNF, B256, 0, stream>>>(out, 0.0f, (int)NF);
        fill_f32<<<(NNODES * H + 255) / 256, B256, 0, stream>>>(denom, 0.0f, NNODES * H);
        fill_u32<<<(NNODES * H + 255) / 256, B256, 0, stream>>>(amax, ENC_NEG_INF, NNODES * H);

        // 3) Attention logits + segment max
        edge_alpha<<<(NEDGES * H + 255) / 256, B256, 0, stream>>>(
            srcI, dstI, q, k, alpha, amax, H, rsqrtf((float)hd));

        // 4) Softmax numerators, denominators, weighted message aggregation
        edge_msg<<<gEdgeWave, B256, 0, stream>>>(srcI, dstI, v, alpha, amax, denom, out, H);

        // 5) Normalize + skip + ReLU
        finalize_node<<<gNF, B256, 0, stream>>>(s, denom, out, H);

        in = out;
    }

    // Global mean pool
    fill_f32<<<(NGRAPH * DMODEL + 255) / 256, B256, 0, stream>>>(psum, 0.0f, NGRAPH * DMODEL);
    fill_f32<<<1, NGRAPH, 0, stream>>>(pcnt, 0.0f, NGRAPH);
    pool_acc<<<gNF, B256, 0, stream>>>(in, batch, psum, pcnt);
    pool_div<<<(NGRAPH * DMODEL + 255) / 256, B256, 0, stream>>>(psum, pcnt, (float*)d_out);
}